// Hebbian_30296699306632
// MI455X (gfx1250) — compile-verified
//
#include <hip/hip_runtime.h>

typedef __attribute__((ext_vector_type(2))) float v2f;
typedef __attribute__((ext_vector_type(8))) float v8f;

#define DDIM 2048
#define LMBDA 1.0f
#define ETA 0.005f

// ---------------------------------------------------------------------------
// Kernel 1: per 16-row/col strip, compute
//   m[j0+jj] = (1/D) * sum_k x[j0+jj,k] * x[k,j0+jj]   (diag of strip GEMM)
//   r[j0+mm] = (1/D) * sum_k x[j0+mm,k]                (rowsum via ones-B)
// using V_WMMA_F32_16X16X4_F32 (fp32 in, fp32 acc — no precision loss).
// 8 waves split the k range; partials reduced in fixed order via LDS.
// ---------------------------------------------------------------------------
__global__ __launch_bounds__(256) void hebb_reduce(const float* __restrict__ x,
                                                   float* __restrict__ m_out,
                                                   float* __restrict__ r_out) {
  __shared__ float lds_m[8 * 16];
  __shared__ float lds_r[8 * 16];

  const int j0   = blockIdx.x * 16;
  const int tid  = threadIdx.x;
  const int lane = tid & 31;
  const int wave = tid >> 5;

  // 32-bit A 16x4 layout: lane l -> row (l&15), K base 2*(l>>4), float2 {K, K+1}
  const int mrow  = lane & 15;
  const int khalf = (lane >> 4) << 1;

  const float*  arow = x + (size_t)(j0 + mrow) * DDIM;  // A: row strip of x
  const size_t  bcol = (size_t)(j0 + mrow);             // B: column strip of x

  v8f cd = {};                      // diag(X_strip * X_colstrip) accumulator
  v8f cr = {};                      // rowsum accumulator (B = ones)
  const v2f ones = {1.0f, 1.0f};

  const int kbeg = wave * (DDIM / 8);
  const int kend = kbeg + (DDIM / 8);

#pragma unroll 4
  for (int k = kbeg; k < kend; k += 4) {
    v2f a = *(const v2f*)(arow + k + khalf);
    v2f b;
    b.x = x[(size_t)(k + khalf + 0) * DDIM + bcol];
    b.y = x[(size_t)(k + khalf + 1) * DDIM + bcol];
    // v_wmma_f32_16x16x4_f32: (neg_a, A, neg_b, B, c_mod, C, reuse_a, reuse_b)
    cd = __builtin_amdgcn_wmma_f32_16x16x4_f32(false, a, false, b,    (short)0, cd, false, false);
    cr = __builtin_amdgcn_wmma_f32_16x16x4_f32(false, a, false, ones, (short)0, cr, false, false);
  }

  // Extract diagonal: C element (M,N): lanes0-15 -> (v, lane); lanes16-31 -> (v+8, lane-16)
  // Diagonal held by lanes 0-7 (cd[lane]) and lanes 24-31 (cd[lane-24]).
  if (lane < 8) {
    lds_m[wave * 16 + lane] = cd[lane];
  } else if (lane >= 24) {
    lds_m[wave * 16 + (lane - 16)] = cd[lane - 24];
  }
  // Row sums: every column of cr equals the row sum; read column N=0
  // (lane 0 holds M=0..7 in cr[0..7], lane 16 holds M=8..15).
  if (lane == 0) {
#pragma unroll
    for (int v = 0; v < 8; ++v) lds_r[wave * 16 + v] = cr[v];
  }
  if (lane == 16) {
#pragma unroll
    for (int v = 0; v < 8; ++v) lds_r[wave * 16 + 8 + v] = cr[v];
  }
  __syncthreads();

  // Deterministic fixed-order reduction of the 8 wave partials.
  if (tid < 16) {
    float ms = 0.0f, rs = 0.0f;
#pragma unroll
    for (int w = 0; w < 8; ++w) {
      ms += lds_m[w * 16 + tid];
      rs += lds_r[w * 16 + tid];
    }
    m_out[j0 + tid] = ms * (1.0f / DDIM);
    r_out[j0 + tid] = rs * (1.0f / DDIM);
  }
}

// ---------------------------------------------------------------------------
// Kernel 2: bandwidth-bound elementwise pass (this is the roofline term).
//   y[i,j]  = LMBDA * kernel[i,j] * m[j]
//   nk[i,j] = kernel[i,j] + ETA * x[i,j] * r[i]
// float4 (b128) loads/stores, 4 elements per thread.
// ---------------------------------------------------------------------------
__global__ __launch_bounds__(256) void hebb_elem(const float* __restrict__ x,
                                                 const float* __restrict__ kern,
                                                 const float* __restrict__ m,
                                                 const float* __restrict__ r,
                                                 float* __restrict__ y,
                                                 float* __restrict__ nk) {
  const size_t idx = ((size_t)blockIdx.x * blockDim.x + threadIdx.x) << 2;
  const int row = (int)(idx >> 11);          // / 2048
  const int col = (int)(idx & (DDIM - 1));   // col, multiple of 4 -> aligned

  float4 k4 = *(const float4*)(kern + idx);
  float4 x4 = *(const float4*)(x + idx);
  float4 m4 = *(const float4*)(m + col);
  const float rb = r[row] * ETA;

  float4 y4, n4;
  y4.x = LMBDA * k4.x * m4.x;  n4.x = fmaf(x4.x, rb, k4.x);
  y4.y = LMBDA * k4.y * m4.y;  n4.y = fmaf(x4.y, rb, k4.y);
  y4.z = LMBDA * k4.z * m4.z;  n4.z = fmaf(x4.z, rb, k4.z);
  y4.w = LMBDA * k4.w * m4.w;  n4.w = fmaf(x4.w, rb, k4.w);

  *(float4*)(y + idx)  = y4;
  *(float4*)(nk + idx) = n4;
}

extern "C" void kernel_launch(void* const* d_in, const int* in_sizes, int n_in,
                              void* d_out, int out_size, void* d_ws, size_t ws_size,
                              hipStream_t stream) {
  (void)in_sizes; (void)n_in; (void)out_size; (void)ws_size;
  const float* x    = (const float*)d_in[0];
  const float* kern = (const float*)d_in[1];

  float* y  = (float*)d_out;                         // first D*D floats
  float* nk = y + (size_t)DDIM * DDIM;               // second D*D floats

  float* m = (float*)d_ws;                           // D floats
  float* r = m + DDIM;                               // D floats

  hipLaunchKernelGGL(hebb_reduce, dim3(DDIM / 16), dim3(256), 0, stream, x, m, r);

  const size_t total = (size_t)DDIM * DDIM / 4;      // 4 elems / thread
  hipLaunchKernelGGL(hebb_elem, dim3((unsigned)(total / 256)), dim3(256), 0, stream,
                     x, kern, m, r, y, nk);
}